// Contrast_loss_33114197852528
// MI455X (gfx1250) — compile-verified
//
#include <hip/hip_runtime.h>
#include <hip/hip_bf16.h>

// loss = 0.5*||S||^2 - 1.5*Sum_L ||S_L||^2 + Sum_i ||f_i||^2
//   S_L = sum of feat rows with label L (L in [0,100)), S = sum_L S_L.
//
// Workspace layout (floats):
//   ws[0 .. 128*256)      : sl  -- per-label sums, padded to 128 rows (rows 100..127 stay 0)
//   ws[128*256 .. +8)     : acc -- acc[0] = Sum_i ||f_i||^2

typedef __attribute__((ext_vector_type(2))) float v2f;
typedef __attribute__((ext_vector_type(8))) float v8f;

#define B_ROWS 8192
#define D_COLS 256
#define NLAB   100
#define SL_ROWS 128  // padded for 8 x 16 WMMA tiles

__global__ void init_ws_kernel(float* __restrict__ ws, int n) {
    int i = blockIdx.x * blockDim.x + threadIdx.x;
    if (i < n) ws[i] = 0.0f;
}

// 32 blocks x 256 threads. Block (rg, cg) handles rows [rg*1024, rg*1024+1024)
// and columns [cg*64, cg*64+64). Per-label partials staged in LDS (ds_add_f32),
// flushed to global sl with f32 atomics. Also accumulates Sum ||f_i||^2.
__global__ void label_accum_kernel(const float* __restrict__ feat,
                                   const int*   __restrict__ label,
                                   float* __restrict__ sl,
                                   float* __restrict__ acc) {
    __shared__ float lsum[NLAB * 64];   // 25.6 KB of the 320 KB WGP LDS
    const int tid = threadIdx.x;
    const int bx  = blockIdx.x;         // 0..31
    const int rg  = bx >> 2;            // 0..7  (row group of 1024 rows)
    const int cg  = bx & 3;             // 0..3  (column group of 64 cols)
    const int cl  = tid & 63;           // local column
    const int ro  = tid >> 6;           // 0..3  (row sub-chunk)

    for (int i = tid; i < NLAB * 64; i += 256) lsum[i] = 0.0f;
    __syncthreads();

    const int c = cg * 64 + cl;
    const int rbase = rg * 1024 + ro * 256;
    float ssq = 0.0f;
    for (int i = 0; i < 256; ++i) {
        const int r = rbase + i;
        const float v = feat[r * D_COLS + c];
        const int lb = label[r];                 // labels in [0,100)
        atomicAdd(&lsum[lb * 64 + cl], v);       // LDS ds_add_f32
        ssq += v * v;
    }
    __syncthreads();

    for (int i = tid; i < NLAB * 64; i += 256) {
        const int L = i >> 6, cc = i & 63;
        atomicAdd(&sl[L * D_COLS + cg * 64 + cc], lsum[i]);
    }
    atomicAdd(acc, ssq);
}

// One block of 256 threads = 8 wave32s.
//  * term_all  = ||sum_L S_L||^2 via per-column reduction + tree reduce.
//  * term_same = trace(M M^T) for the padded 128x256 matrix M=sl: wave w computes
//    the diagonal 16x16 tile (rows 16w..16w+15) with V_WMMA_F32_16X16X4_F32,
//    K-loop over 256 in steps of 4, then extracts the C-matrix diagonal.
__global__ void finalize_kernel(const float* __restrict__ sl,
                                const float* __restrict__ acc,
                                float* __restrict__ out) {
    __shared__ float red[256];
    __shared__ float s_same;
    const int tid = threadIdx.x;
    if (tid == 0) s_same = 0.0f;

    // ---- term_all: S[d] = sum over the 100 label rows; reduce S[d]^2 ----
    float s = 0.0f;
    for (int L = 0; L < NLAB; ++L) s += sl[L * D_COLS + tid];
    red[tid] = s * s;
    __syncthreads();
    for (int off = 128; off > 0; off >>= 1) {
        if (tid < off) red[tid] += red[tid + off];
        __syncthreads();
    }
    const float term_all = red[0];   // valid in all threads after the barriers
    __syncthreads();

    // ---- term_same: diagonal tiles of sl * sl^T via f32 WMMA ----
    const int wave = tid >> 5;          // 0..7 -> diagonal tile index
    const int lane = tid & 31;
    const int m    = lane & 15;         // A: M index / B: N index for this lane
    const int kh   = lane >> 4;         // half-wave selects K pair

    const float* tile = sl + (wave * 16) * D_COLS;   // rows 16w..16w+15
    v8f cacc = {};
    for (int k0 = 0; k0 < D_COLS; k0 += 4) {
        v2f a, b;
        // A (16x4): lanes 0-15 hold M=0..15 with K={0,1} in v0,v1; lanes 16-31 K={2,3}
        a.x = tile[m * D_COLS + k0 + 2 * kh + 0];
        a.y = tile[m * D_COLS + k0 + 2 * kh + 1];
        // B (4x16) = tile^T slice: row K striped across lanes within a VGPR
        //   v0: K=kh (0 for lanes 0-15, 1 for lanes 16-31), v1: K=2+kh; N = lane&15
        b.x = tile[m * D_COLS + k0 + kh];
        b.y = tile[m * D_COLS + k0 + 2 + kh];
        cacc = __builtin_amdgcn_wmma_f32_16x16x4_f32(
            /*neg_a=*/false, a, /*neg_b=*/false, b,
            /*c_mod=*/(short)0, cacc, /*reuse_a=*/false, /*reuse_b=*/false);
    }
    // C/D layout: VGPR v holds M=v (lanes 0-15, N=lane) and M=v+8 (lanes 16-31,
    // N=lane-16). Diagonal (M==N): lanes 0..7 take c[lane]; lanes 24..31 take c[lane-24].
    float local = 0.0f;
#pragma unroll
    for (int v = 0; v < 8; ++v) {
        const bool take = (lane < 8 && lane == v) || (lane >= 24 && (lane - 24) == v);
        local += take ? cacc[v] : 0.0f;
    }
    atomicAdd(&s_same, local);          // LDS ds_add_f32
    __syncthreads();

    if (tid == 0) {
        out[0] = 0.5f * term_all - 1.5f * s_same + acc[0];
    }
}

extern "C" void kernel_launch(void* const* d_in, const int* in_sizes, int n_in,
                              void* d_out, int out_size, void* d_ws, size_t ws_size,
                              hipStream_t stream) {
    const float* feat  = (const float*)d_in[0];   // [8192, 256] f32
    const int*   label = (const int*)d_in[1];     // [8192]
    float* out = (float*)d_out;                   // scalar loss

    float* sl  = (float*)d_ws;                    // 128*256 floats
    float* acc = sl + SL_ROWS * D_COLS;           // 8 floats

    const int ws_floats = SL_ROWS * D_COLS + 8;
    init_ws_kernel<<<(ws_floats + 255) / 256, 256, 0, stream>>>(sl, ws_floats);
    label_accum_kernel<<<32, 256, 0, stream>>>(feat, label, sl, acc);
    finalize_kernel<<<1, 256, 0, stream>>>(sl, acc, out);
}